// TreeScorer_45767171506842
// MI455X (gfx1250) — compile-verified
//
#include <hip/hip_runtime.h>
#include <hip/hip_bf16.h>

typedef __attribute__((ext_vector_type(2))) float v2f;
typedef __attribute__((ext_vector_type(8))) float v8f;

#define BB 8
#define SS 128
#define HH 768
#define NSPANS 8255   // S*(S+1)/2 - 1

// ---------------------------------------------------------------------------
// Kernel 1: proj[r][0:4] = [tok[r]·W[:H,0], tok[r]·W[:H,1],
//                           tok[r]·W[H:,0], tok[r]·W[H:,1]]
// over r = b*S+s (1024 rows).  One wave per 16-row tile, exact fp32 WMMA
// V_WMMA_F32_16X16X4_F32, K-loop of 192 steps.
// Lanes n>=4 compute garbage columns of D (from valid-but-arbitrary W reads);
// those columns are never stored, so no masking / exec divergence is needed.
// ---------------------------------------------------------------------------
__global__ __launch_bounds__(32) void proj_wmma_kernel(
    const float* __restrict__ tok, const float* __restrict__ W,
    float* __restrict__ proj) {
  const int lane = threadIdx.x;             // 0..31 (wave32)
  const int rowBase = blockIdx.x * 16;      // 64 tiles over 1024 rows
  const int m = lane & 15;                  // A-matrix row in tile
  const int n = lane & 15;                  // B/D column
  const int koff = (lane >> 4) << 1;        // 0 (lanes 0-15) or 2 (lanes 16-31)
  // Wcat columns: 0 -> W[:H,0], 1 -> W[:H,1], 2 -> W[H:,0], 3 -> W[H:,1].
  // For n>=4 these aliases stay in-bounds; the results land in dead D columns.
  const int col  = n & 1;
  const int rofs = (n & 2) ? HH : 0;
  const float* wcol = W + (size_t)rofs * 2 + col;   // element k at wcol[2*k]

  const float* arow = tok + (size_t)(rowBase + m) * HH;
  v8f c = {};

#pragma unroll 4
  for (int k = 0; k < HH; k += 4) {
    const int kk = k + koff;
    // A 16x4 fp32 layout: VGPR0 = K=koff, VGPR1 = K=koff+1 per lane-half
    float2 av = *(const float2*)(arow + kk);
    v2f a; a.x = av.x; a.y = av.y;
    // B 4x16: row K striped across lanes (VGPR0 rows {0,2}, VGPR1 rows {1,3})
    v2f bm;
    bm.x = wcol[2 * (size_t)kk];
    bm.y = wcol[2 * (size_t)(kk + 1)];
    c = __builtin_amdgcn_wmma_f32_16x16x4_f32(
        /*neg_a=*/false, a, /*neg_b=*/false, bm,
        /*c_mod=*/(short)0, c, /*reuse_a=*/false, /*reuse_b=*/false);
  }

  // D 16x16 f32: VGPR v -> M = v (lanes 0-15) / v+8 (lanes 16-31), N = lane&15
  if (n < 4) {
    const int rAdd = (lane >> 4) * 8;
#pragma unroll
    for (int v = 0; v < 8; ++v)
      proj[(size_t)(rowBase + rAdd + v) * 4 + n] = c[v];
  }
}

// ---------------------------------------------------------------------------
// Kernel 2: per-span 2-class softmax.
// Span groups: g = 0..126, size k = 127-g, count g+2, i = g*(g+3)/2 + t.
// Writes probs directly into d_out[(b*N+i)*2 + {0,1}].
// ---------------------------------------------------------------------------
__global__ __launch_bounds__(128) void probs_kernel(
    const float* __restrict__ proj, const float* __restrict__ bias,
    float* __restrict__ out_probs) {
  const int g = blockIdx.x;   // 0..126
  const int b = blockIdx.y;   // 0..7
  const int t = threadIdx.x;  // 0..127
  if (t >= g + 2) return;

  const int i = g * (g + 3) / 2 + t;
  const int s = t;
  const int e = t + (127 - g);           // end (exclusive), e-1 = last token

  const float* pf = proj + ((size_t)b * SS + s) * 4;
  const float* pl = proj + ((size_t)b * SS + (e - 1)) * 4;
  const float l0 = pf[0] + pl[2] + bias[0];
  const float l1 = pf[1] + pl[3] + bias[1];

  const float mx = fmaxf(l0, l1);
  const float e0 = expf(l0 - mx), e1 = expf(l1 - mx);
  const float inv = 1.0f / (e0 + e1);

  float* o = out_probs + ((size_t)b * NSPANS + i) * 2;
  o[0] = e0 * inv;
  o[1] = e1 * inv;
}

// ---------------------------------------------------------------------------
// Kernel 3: greedy non-overlap selection.  Spans are processed in
// non-increasing size order, so the selected set is always a set of disjoint
// intervals, and (ov & sel) == {cover[s], cover[e-1]}.  O(1) per step plus
// O(S) per take, instead of the reference's O(N^2) scan.
// One block per batch; thread 0 runs the serial scan over LDS.
// scoreSlot[i] = is_slot ? p1 : -p1  (selected spans are always slots => >0).
// ---------------------------------------------------------------------------
__global__ __launch_bounds__(128) void select_kernel(
    const float* __restrict__ out_probs, float* __restrict__ out_sel) {
  __shared__ float scoreSlot[NSPANS];
  __shared__ unsigned char selB[NSPANS];
  __shared__ int cover[SS];

  const int b = blockIdx.x;
  const int t = threadIdx.x;
  const float* pb = out_probs + (size_t)b * NSPANS * 2;

  for (int i = t; i < NSPANS; i += 128) {
    const float p0 = pb[2 * i], p1 = pb[2 * i + 1];
    scoreSlot[i] = (p1 > p0) ? p1 : -p1;
    selB[i] = 0;
  }
  if (t < SS) cover[t] = -1;
  __syncthreads();

  if (t == 0) {
    int g = 0, st = 0;   // group / start, advanced incrementally
    for (int i = 0; i < NSPANS; ++i) {
      const int s = st;
      const int e = st + (127 - g);
      const float sc = scoreSlot[i];
      if (sc > 0.f) {                       // is_slot
        const int j1 = cover[s];
        const int j2 = cover[e - 1];
        const bool ok = (j1 < 0 || sc > scoreSlot[j1]) &&
                        (j2 < 0 || sc > scoreSlot[j2]);
        if (ok) {
          if (j1 >= 0) selB[j1] = 0;
          if (j2 >= 0) selB[j2] = 0;
          if (j1 >= 0 || j2 >= 0) {
            for (int p = 0; p < SS; ++p) {
              const int c = cover[p];
              if (c >= 0 && (c == j1 || c == j2)) cover[p] = -1;
            }
          }
          for (int p = s; p < e; ++p) cover[p] = i;
          selB[i] = 1;
        }
      }
      // advance span coordinates
      ++st;
      if (st == g + 2) { st = 0; ++g; }
    }
  }
  __syncthreads();

  float* os = out_sel + (size_t)b * NSPANS;
  for (int i = t; i < NSPANS; i += 128)
    os[i] = selB[i] ? 1.0f : 0.0f;
}

// ---------------------------------------------------------------------------
extern "C" void kernel_launch(void* const* d_in, const int* in_sizes, int n_in,
                              void* d_out, int out_size, void* d_ws, size_t ws_size,
                              hipStream_t stream) {
  const float* tok  = (const float*)d_in[0];  // (B, S, H) fp32
  const float* W    = (const float*)d_in[1];  // (2H, 2)  fp32
  const float* bias = (const float*)d_in[2];  // (2,)     fp32

  float* out_probs = (float*)d_out;                              // B*N*2
  float* out_sel   = (float*)d_out + (size_t)BB * NSPANS * 2;    // B*N
  float* proj      = (float*)d_ws;                               // 1024*4 floats

  // 1) per-token projections via fp32 WMMA (exact)
  proj_wmma_kernel<<<(BB * SS) / 16, 32, 0, stream>>>(tok, W, proj);

  // 2) per-span softmax -> probs in d_out
  probs_kernel<<<dim3(127, BB), 128, 0, stream>>>(proj, bias, out_probs);

  // 3) greedy selection -> selected flags (as 0/1 floats) in d_out
  select_kernel<<<BB, 128, 0, stream>>>(out_probs, out_sel);
}